// LocallyConnected3D_14817637171811
// MI455X (gfx1250) — compile-verified
//
#include <hip/hip_runtime.h>
#include <hip/hip_bf16.h>

// LocallyConnected3D on MI455X (gfx1250, wave32).
//
// Bandwidth-bound: the 174 MB of per-location weights dominate (inputs 4 MB sit
// in the 192 MB L2). One wave per output location computes the (B=8..16) x
// (F=16) tile with 54 x V_WMMA_F32_16X16X4_F32 (exact fp32), streaming each
// weight row as fully coalesced 64B per 16-lane half.

typedef float v2f __attribute__((ext_vector_type(2)));
typedef float v8f __attribute__((ext_vector_type(8)));

#define B_    8
#define X_    32
#define Y_    32
#define Z_    16
#define CIN_  8
#define KS    3
#define F_    16
#define OX    30
#define OY    30
#define OZ    14
#define NLOC  (OX * OY * OZ)          // 12600
// input strides (floats), layout (B, X, Y, Z, CIN)
#define IN_B  (X_ * Y_ * Z_ * CIN_)   // 131072
#define IN_X  (Y_ * Z_ * CIN_)        // 4096
#define IN_Y  (Z_ * CIN_)             // 128
#define IN_Z  (CIN_)                  // 8
// output strides (floats), layout (B, OX, OY, OZ, F)
#define OUT_B (NLOC * F_)             // 201600
#define OUT_X (OY * OZ * F_)          // 6720
#define OUT_Y (OZ * F_)               // 224
#define OUT_Z (F_)                    // 16
// weights: (OX,OY,OZ, K,K,K, CIN, F) -> per-location (216 x 16) row-major
#define WLOC  (KS * KS * KS * CIN_ * F_) // 3456

__global__ __launch_bounds__(256) void
lc3d_wmma_f32(const float* __restrict__ in,
              const float* __restrict__ wgt,
              const float* __restrict__ bias,
              float* __restrict__ out) {
  const int lane = threadIdx.x & 31;
  const int wave = threadIdx.x >> 5;
  const int loc  = blockIdx.x * 8 + wave;        // grid sized exactly: < 12600

  const int x   = loc / (OY * OZ);
  const int rem = loc - x * (OY * OZ);
  const int y   = rem / OZ;
  const int z   = rem - y * OZ;

  const int n    = lane & 15;   // N column = filter f
  const int half = lane >> 4;   // 0 -> K pair {0,1}; 1 -> K pair {2,3}
  const int bm   = lane & 7;    // batch row (lanes 8..15 duplicate, masked to 0)
  const float mask = ((lane & 15) < 8) ? 1.0f : 0.0f;

  // Weight pointer for this lane: row (4t + half*2), column n.
  const float* wr = wgt + (size_t)loc * WLOC + (size_t)(half * 2) * F_ + n;
  // Input base for this lane's batch row, at channel offset half*2.
  const float* ib = in + (size_t)bm * IN_B + (size_t)x * IN_X +
                    (size_t)y * IN_Y + (size_t)z * IN_Z + half * 2;

  v8f c = {0.f, 0.f, 0.f, 0.f, 0.f, 0.f, 0.f, 0.f};

  #pragma unroll
  for (int i = 0; i < KS; ++i) {
    #pragma unroll
    for (int j = 0; j < KS; ++j) {
      #pragma unroll
      for (int k = 0; k < KS; ++k) {
        const float* ip = ib + i * IN_X + j * IN_Y + k * IN_Z;
        #pragma unroll
        for (int cc = 0; cc < 2; ++cc) {     // channel group c0 = cc*4
          // A 16x4 f32: lane holds K = half*2 + {0,1} -> contiguous float2.
          float2 av = *(const float2*)(ip + cc * 4);
          v2f a;
          a.x = av.x * mask;                 // zero rows M=8..15
          a.y = av.y * mask;
          // B 4x16 f32: VGPR0 = row (half*2), VGPR1 = row (half*2 + 1).
          v2f b;
          b.x = wr[0];
          b.y = wr[F_];
          c = __builtin_amdgcn_wmma_f32_16x16x4_f32(
              /*neg_a=*/false, a, /*neg_b=*/false, b,
              /*c_mod=*/(short)0, c, /*reuse_a=*/false, /*reuse_b=*/false);
          wr += 4 * F_;                      // advance 4 K-rows
        }
      }
    }
  }

  // C/D 16x16 f32 layout: VGPR r, lanes 0..15 -> row M=r, col N=lane.
  // Only rows 0..7 (real batches) are stored; each store writes 16 contiguous
  // floats (f = 0..15) per batch -> fully coalesced.
  const float bval = bias[loc * F_ + n];
  if (lane < 16) {
    float* op = out + (size_t)x * OUT_X + (size_t)y * OUT_Y +
                (size_t)z * OUT_Z + n;
    #pragma unroll
    for (int r = 0; r < 8; ++r) {
      op[(size_t)r * OUT_B] = c[r] + bval;
    }
  }
}

extern "C" void kernel_launch(void* const* d_in, const int* in_sizes, int n_in,
                              void* d_out, int out_size, void* d_ws, size_t ws_size,
                              hipStream_t stream) {
  const float* in   = (const float*)d_in[0];
  const float* wgt  = (const float*)d_in[1];
  const float* bias = (const float*)d_in[2];
  float* out        = (float*)d_out;

  // 12600 locations, 8 waves (256 threads) per block -> exactly 1575 blocks.
  dim3 grid(NLOC / 8);
  dim3 block(256);
  lc3d_wmma_f32<<<grid, block, 0, stream>>>(in, wgt, bias, out);
}